// FuzzyModel_47296179863578
// MI455X (gfx1250) — compile-verified
//
#include <hip/hip_runtime.h>
#include <math.h>

// ---------------- problem constants (from reference) ----------------
constexpr int NB = 8, CH = 3, H = 512, W = 512;   // x: (8,3,512,512)
constexpr int HD = 256, WD = 256;                 // y: (8,3,256,256)
constexpr int NG = 5, NM = 3;                     // G gaussians, M membership fns
// workspace layout (floats): [0,YN) = downsampled y; [YN, YN+CTOT) = constants
constexpr size_t YN  = (size_t)NB * CH * HD * WD; // 1,572,864 floats (6 MB)
constexpr int CW   = 0;          // folded weight matrix, 16 rows x 16 cols (K x N)
constexpr int CMFC = 256;        // mf centers (15 used)
constexpr int CSL  = 256 + 16;   // softplus(mf_sL)+eps
constexpr int CSR  = 256 + 32;   // softplus(mf_sR)+eps
constexpr int CTOT = 256 + 48;

typedef __attribute__((ext_vector_type(2))) float v2f;
typedef __attribute__((ext_vector_type(8))) float v8f;

__device__ __forceinline__ float softplus_f(float x) {
  return (x > 20.0f) ? x : log1pf(expf(x));
}

__device__ __forceinline__ void make_gauss(float g[7]) {
  float s = 0.0f;
#pragma unroll
  for (int i = 0; i < 7; ++i) {
    float r = (float)(i - 3);
    g[i] = expf(-r * r * (1.0f / (2.0f * 0.9f * 0.9f)));
    s += g[i];
  }
  float inv = 1.0f / s;
#pragma unroll
  for (int i = 0; i < 7; ++i) g[i] *= inv;
}

__device__ __forceinline__ int reflect_idx(int i, int n) {
  if (i < 0) i = -i;
  if (i >= n) i = 2 * n - 2 - i;
  return i;
}

// ---------------- K0: fold GMM parameters into a 12x5 weight matrix ----------------
// logits[g] = W[:,g] . (1, px0,px1,px2, px0^2,px1^2,px2^2, px0*px1, px0*px2, px1*px2)
__global__ void precompute_kernel(const float* __restrict__ mu,
                                  const float* __restrict__ Lraw,
                                  const float* __restrict__ pi,
                                  const float* __restrict__ mfc,
                                  const float* __restrict__ msl,
                                  const float* __restrict__ msr,
                                  float* __restrict__ cst) {
  if (threadIdx.x != 0 || blockIdx.x != 0) return;
  for (int i = 0; i < CTOT; ++i) cst[i] = 0.0f;

  // log_softmax(pi)
  float mx = pi[0];
  for (int g = 1; g < NG; ++g) mx = fmaxf(mx, pi[g]);
  float se = 0.0f;
  for (int g = 0; g < NG; ++g) se += expf(pi[g] - mx);
  float lse = mx + logf(se);

  for (int g = 0; g < NG; ++g) {
    float L[3][3];
    for (int r = 0; r < 3; ++r)
      for (int c = 0; c < 3; ++c)
        L[r][c] = (c <= r) ? Lraw[(g * 3 + r) * 3 + c] : 0.0f;
    for (int r = 0; r < 3; ++r) {
      float d = softplus_f(L[r][r]) + 1e-4f;   // MIN_DIAG
      d = softplus_f(d) + 1e-4f;               // SIG_EPS
      L[r][r] = d;
    }
    float S[3][3];
    for (int r = 0; r < 3; ++r)
      for (int c = 0; c < 3; ++c) {
        float s = 0.0f;
        for (int k = 0; k < 3; ++k) s += L[r][k] * L[c][k];
        if (r == c) s += 1e-4f;                // + SIG_EPS * I
        S[r][c] = s;
      }
    float detS = S[0][0] * (S[1][1] * S[2][2] - S[1][2] * S[2][1])
               - S[0][1] * (S[1][0] * S[2][2] - S[1][2] * S[2][0])
               + S[0][2] * (S[1][0] * S[2][1] - S[1][1] * S[2][0]);
    float logdet = logf(detS);

    float A[3][3];
    for (int r = 0; r < 3; ++r)
      for (int c = 0; c < 3; ++c)
        A[r][c] = S[r][c] + ((r == c) ? 1e-3f : 0.0f);  // + JITTER * I
    float detA = A[0][0] * (A[1][1] * A[2][2] - A[1][2] * A[2][1])
               - A[0][1] * (A[1][0] * A[2][2] - A[1][2] * A[2][0])
               + A[0][2] * (A[1][0] * A[2][1] - A[1][1] * A[2][0]);
    float id = 1.0f / detA;
    float inv[3][3];
    inv[0][0] =  (A[1][1] * A[2][2] - A[1][2] * A[2][1]) * id;
    inv[0][1] = -(A[0][1] * A[2][2] - A[0][2] * A[2][1]) * id;
    inv[0][2] =  (A[0][1] * A[1][2] - A[0][2] * A[1][1]) * id;
    inv[1][0] = inv[0][1];
    inv[1][1] =  (A[0][0] * A[2][2] - A[0][2] * A[2][0]) * id;
    inv[1][2] = -(A[0][0] * A[1][2] - A[0][2] * A[1][0]) * id;
    inv[2][0] = inv[0][2];
    inv[2][1] = inv[1][2];
    inv[2][2] =  (A[0][0] * A[1][1] - A[0][1] * A[1][0]) * id;

    float m0 = mu[g * 3 + 0], m1 = mu[g * 3 + 1], m2 = mu[g * 3 + 2];
    float b0 = inv[0][0] * m0 + inv[0][1] * m1 + inv[0][2] * m2;
    float b1 = inv[1][0] * m0 + inv[1][1] * m1 + inv[1][2] * m2;
    float b2 = inv[2][0] * m0 + inv[2][1] * m1 + inv[2][2] * m2;
    float cq = m0 * b0 + m1 * b1 + m2 * b2;

    cst[CW + 0 * 16 + g] = -0.5f * cq - 0.5f * logdet + (pi[g] - lse);
    cst[CW + 1 * 16 + g] = b0;
    cst[CW + 2 * 16 + g] = b1;
    cst[CW + 3 * 16 + g] = b2;
    cst[CW + 4 * 16 + g] = -0.5f * inv[0][0];
    cst[CW + 5 * 16 + g] = -0.5f * inv[1][1];
    cst[CW + 6 * 16 + g] = -0.5f * inv[2][2];
    cst[CW + 7 * 16 + g] = -inv[0][1];
    cst[CW + 8 * 16 + g] = -inv[0][2];
    cst[CW + 9 * 16 + g] = -inv[1][2];

    for (int m = 0; m < NM; ++m) {
      cst[CMFC + g * 3 + m] = mfc[g * 3 + m];
      cst[CSL  + g * 3 + m] = softplus_f(msl[g * 3 + m]) + 1e-12f;
      cst[CSR  + g * 3 + m] = softplus_f(msr[g * 3 + m]) + 1e-12f;
    }
  }
}

// ---------------- K1: 7x7 Gaussian blur, reflect pad, stride 2 ----------------
__global__ __launch_bounds__(256) void down_kernel(const float* __restrict__ x,
                                                   float* __restrict__ y) {
  int t = blockIdx.x * blockDim.x + threadIdx.x;
  float gk[7];
  make_gauss(gk);
  int v  = t % WD;
  int u  = (t / WD) % HD;
  int nc = t / (WD * HD);
  const float* xp = x + (size_t)nc * H * W;
  float acc = 0.0f;
#pragma unroll
  for (int ti = 0; ti < 7; ++ti) {
    int ii = reflect_idx(2 * u + ti - 3, H);
    float rw = gk[ti];
    const float* row = xp + (size_t)ii * W;
#pragma unroll
    for (int si = 0; si < 7; ++si) {
      int jj = reflect_idx(2 * v + si - 3, W);
      acc += rw * gk[si] * row[jj];
    }
  }
  y[t] = acc;
}

// ---------------- K2: fused upsample + WMMA logits + dual softmax ----------------
// Per wave: 32 pixels. Features staged in LDS, two 16-pixel A tiles,
// 3x V_WMMA_F32_16X16X4_F32 each (K = 12), D staged via LDS for per-lane epilogue.
__global__ __launch_bounds__(256) void fuzzy_main(const float* __restrict__ y,
                                                  const float* __restrict__ cst,
                                                  float* __restrict__ out) {
  __shared__ float sFeat[8 * 32 * 12];  // [wave][pixel-in-wave][feature]
  __shared__ float sD[8 * 512];         // [wave][tile*256 + m*16 + n]
  const int lane = threadIdx.x & 31;
  const int wave = threadIdx.x >> 5;
  const long base = ((long)blockIdx.x * 8 + wave) * 32;
  const long p = base + lane;

  const int j = (int)(p % W);
  const int i = (int)((p / W) % H);
  const int n = (int)(p / ((long)W * H));

  float gk[7];
  make_gauss(gk);

  // transposed-conv reconstruction: xr[i,j] = 4 * sum g[t]g[s] * y[(i+t-3)/2, (j+s-3)/2]
  // over even (i+t-3),(j+s-3) inside [0, 2*HD)
  float px[3];
#pragma unroll
  for (int c = 0; c < CH; ++c) {
    const float* yc = y + (size_t)(n * CH + c) * HD * WD;
    float acc = 0.0f;
#pragma unroll
    for (int ti = 0; ti < 7; ++ti) {
      int a = i + ti - 3;
      if (a & 1) continue;
      int ya = a >> 1;
      if ((unsigned)ya >= (unsigned)HD) continue;
      float rw = gk[ti];
      const float* row = yc + (size_t)ya * WD;
#pragma unroll
      for (int si = 0; si < 7; ++si) {
        int b = j + si - 3;
        if (b & 1) continue;
        int yb = b >> 1;
        if ((unsigned)yb >= (unsigned)WD) continue;
        acc += rw * gk[si] * row[yb];
      }
    }
    px[c] = 4.0f * acc;
  }

  // quadratic feature vector (K = 12, rows 10/11 zero-padded)
  float* fw = &sFeat[(wave * 32 + lane) * 12];
  fw[0] = 1.0f;
  fw[1] = px[0]; fw[2] = px[1]; fw[3] = px[2];
  fw[4] = px[0] * px[0]; fw[5] = px[1] * px[1]; fw[6] = px[2] * px[2];
  fw[7] = px[0] * px[1]; fw[8] = px[0] * px[2]; fw[9] = px[1] * px[2];
  fw[10] = 0.0f; fw[11] = 0.0f;

  // B (K x N = 4x16 per chunk): VGPR r, lanes 0-15 -> K=r, lanes 16-31 -> K=r+2
  const int ncol  = lane & 15;
  const int khalf = (lane >> 4) * 2;
  v2f B0, B1, B2;
  B0[0] = cst[CW + (0 + khalf) * 16 + ncol]; B0[1] = cst[CW + (1 + khalf) * 16 + ncol];
  B1[0] = cst[CW + (4 + khalf) * 16 + ncol]; B1[1] = cst[CW + (5 + khalf) * 16 + ncol];
  B2[0] = cst[CW + (8 + khalf) * 16 + ncol]; B2[1] = cst[CW + (9 + khalf) * 16 + ncol];

#pragma unroll
  for (int tile = 0; tile < 2; ++tile) {
    const float* ft = &sFeat[(wave * 32 + tile * 16 + (lane & 15)) * 12];
    // A (M x K = 16x4): lanes 0-15 hold K0/K1, lanes 16-31 hold K2/K3
    v2f A0, A1, A2;
    A0[0] = ft[0 + khalf]; A0[1] = ft[1 + khalf];
    A1[0] = ft[4 + khalf]; A1[1] = ft[5 + khalf];
    A2[0] = ft[8 + khalf]; A2[1] = ft[9 + khalf];
    v8f acc8 = {};
    acc8 = __builtin_amdgcn_wmma_f32_16x16x4_f32(false, A0, false, B0, (short)0, acc8, false, false);
    acc8 = __builtin_amdgcn_wmma_f32_16x16x4_f32(false, A1, false, B1, (short)0, acc8, false, false);
    acc8 = __builtin_amdgcn_wmma_f32_16x16x4_f32(false, A2, false, B2, (short)0, acc8, false, false);
    // D: VGPR v -> M = v + 8*(lane>=16), N = lane&15
#pragma unroll
    for (int v = 0; v < 8; ++v) {
      int m = v + (lane >> 4) * 8;
      sD[wave * 512 + tile * 256 + m * 16 + ncol] = acc8[v];
    }
  }

  // epilogue: lane l owns pixel base+l; its logits live at sD[wave*512 + l*16 + g]
  float lg[NG];
#pragma unroll
  for (int g = 0; g < NG; ++g) lg[g] = sD[wave * 512 + lane * 16 + g];
  float mx = lg[0];
#pragma unroll
  for (int g = 1; g < NG; ++g) mx = fmaxf(mx, lg[g]);
  float se = 0.0f;
  float eg[NG];
#pragma unroll
  for (int g = 0; g < NG; ++g) { eg[g] = expf(lg[g] - mx); se += eg[g]; }
  float inv_se = 1.0f / se;

  float* op = out + (size_t)p * (NG * NM);
#pragma unroll
  for (int g = 0; g < NG; ++g) {
    float r = eg[g] * inv_se;  // responsibility
    float z[NM], zm = -1e30f;
#pragma unroll
    for (int m = 0; m < NM; ++m) {
      float cc = cst[CMFC + g * 3 + m];
      float sg = (r < cc) ? cst[CSL + g * 3 + m] : cst[CSR + g * 3 + m];
      float t = (r - cc) / sg;
      z[m] = -0.5f * t * t;
      zm = fmaxf(zm, z[m]);
    }
    float zs = 0.0f, ez[NM];
#pragma unroll
    for (int m = 0; m < NM; ++m) { ez[m] = expf(z[m] - zm); zs += ez[m]; }
    float izs = 1.0f / zs;
#pragma unroll
    for (int m = 0; m < NM; ++m) op[g * 3 + m] = ez[m] * izs;
  }
}

// ---------------- host launcher ----------------
extern "C" void kernel_launch(void* const* d_in, const int* in_sizes, int n_in,
                              void* d_out, int out_size, void* d_ws, size_t ws_size,
                              hipStream_t stream) {
  const float* x    = (const float*)d_in[0];
  const float* mu   = (const float*)d_in[1];
  const float* Lraw = (const float*)d_in[2];
  const float* pi   = (const float*)d_in[3];
  const float* mfc  = (const float*)d_in[4];
  const float* msl  = (const float*)d_in[5];
  const float* msr  = (const float*)d_in[6];
  float* ws  = (float*)d_ws;
  float* y   = ws;
  float* cst = ws + YN;
  float* out = (float*)d_out;

  precompute_kernel<<<1, 32, 0, stream>>>(mu, Lraw, pi, mfc, msl, msr, cst);

  int ydn = (int)(YN / 256);                 // 6144 blocks
  down_kernel<<<ydn, 256, 0, stream>>>(x, y);

  long P = (long)NB * H * W;                 // 2,097,152 pixels
  int blocks = (int)(P / 256);               // 8192 blocks, 8 waves x 32 px each
  fuzzy_main<<<blocks, 256, 0, stream>>>(y, cst, out);
}